// MultiHeadAttention_5102421148328
// MI455X (gfx1250) — compile-verified
//
#include <hip/hip_runtime.h>
#include <hip/hip_bf16.h>
#include <math.h>

typedef float v2f __attribute__((ext_vector_type(2)));
typedef float v8f __attribute__((ext_vector_type(8)));

#define D_MODEL   768
#define S_LEN     4096
#define NUM_HEADS 12
#define D_KK      64

// NT GEMM tiling
#define BM     128         // block M
#define BN     256         // block N
#define KC     16          // K chunk staged in LDS
#define LDSROW 20          // KC + 4 floats pad (80B rows, 16B aligned, bank-spread)

__device__ __forceinline__ void async_copy16(uint32_t lds_byte_off, const float* gptr) {
  // GLOBAL_LOAD_ASYNC_TO_LDS_B128: per-lane 16B global -> LDS, tracked by ASYNCcnt
  asm volatile("global_load_async_to_lds_b128 %0, %1, off"
               :: "v"(lds_byte_off), "v"(gptr) : "memory");
}
__device__ __forceinline__ void wait_async0() {
  asm volatile("s_wait_asynccnt 0x0" ::: "memory");
}

// ---------------------------------------------------------------------------
// NT GEMM (LDS-staged): C[m,n] = scale * sum_k A[m,k] * B[n,k]
// Block 256 threads (8 waves), block tile 128x256, wave grid 2(M)x4(N),
// wave tile 64x64 = 4 M-frags x 4 N-frags of V_WMMA_F32_16X16X4_F32
// (16 accumulators, 2.0 wmma per ds_load_b64).
// A and B K-chunks double-buffered in LDS via global_load_async_to_lds_b128.
// blockIdx.z = head, with per-z element strides sA/sB/sC.
// ---------------------------------------------------------------------------
__global__ void __launch_bounds__(256) gemm_nt_lds(
    const float* __restrict__ A, const float* __restrict__ B,
    float* __restrict__ C,
    int K, int lda, int ldb, int ldc,
    unsigned long long sA, unsigned long long sB, unsigned long long sC,
    float scale)
{
  __shared__ float sm[2 * (BM + BN) * LDSROW];       // [buf][A rows | B rows][LDSROW]
  const uint32_t lds_base = (uint32_t)(uintptr_t)(void*)sm;   // LDS byte offset of sm

  const float* Ab = A + (size_t)blockIdx.z * sA + (size_t)blockIdx.y * BM * lda;
  const float* Bb = B + (size_t)blockIdx.z * sB + (size_t)blockIdx.x * BN * ldb;
  float*       Cb = C + (size_t)blockIdx.z * sC;

  const int tid  = threadIdx.x;
  const int lane = tid & 31;
  const int wave = tid >> 5;
  const int wm   = wave >> 2;                 // 0..1
  const int wn   = wave & 3;                  // 0..3
  const int m0w  = wm * 64;                   // wave M origin within block tile
  const int n0w  = wn * 64;                   // wave N origin within block tile
  const int half = lane >> 4;
  const int l16  = lane & 15;

  // Stage one K-chunk (KC floats per row): A = BM rows, B = BN rows.
  // Row = KC*4 = 64B = 4 b128 segments. A: 512 segs (2/thread), B: 1024 (4/thread).
  auto stage = [&](int buf, int kc) {
    const uint32_t aoff = lds_base + (uint32_t)buf * ((BM + BN) * LDSROW * 4);
    const uint32_t boff = aoff + BM * LDSROW * 4;
    #pragma unroll
    for (int i = 0; i < 2; ++i) {
      const int s   = tid + i * 256;
      const int row = s >> 2;
      const int c   = (s & 3) * 4;
      async_copy16(aoff + (uint32_t)(row * LDSROW + c) * 4, Ab + (size_t)row * lda + kc + c);
    }
    #pragma unroll
    for (int i = 0; i < 4; ++i) {
      const int s   = tid + i * 256;
      const int row = s >> 2;
      const int c   = (s & 3) * 4;
      async_copy16(boff + (uint32_t)(row * LDSROW + c) * 4, Bb + (size_t)row * ldb + kc + c);
    }
  };

  v8f acc[4][4] = {};

  stage(0, 0);
  wait_async0();
  __syncthreads();

  const int nchunk = K / KC;
  for (int c = 0; c < nchunk; ++c) {
    const int buf = c & 1;
    if (c + 1 < nchunk) stage(buf ^ 1, (c + 1) * KC);

    const float* As = sm + buf * ((BM + BN) * LDSROW);
    const float* Bs = As + BM * LDSROW;
    #pragma unroll
    for (int ks = 0; ks < KC; ks += 4) {
      const int col = ks + 2 * half;
      v2f af[4], bf[4];
      #pragma unroll
      for (int i = 0; i < 4; ++i)
        af[i] = *(const v2f*)(As + (m0w + 16 * i + l16) * LDSROW + col);
      #pragma unroll
      for (int j = 0; j < 4; ++j)
        bf[j] = *(const v2f*)(Bs + (n0w + 16 * j + l16) * LDSROW + col);
      #pragma unroll
      for (int i = 0; i < 4; ++i)
        #pragma unroll
        for (int j = 0; j < 4; ++j)
          acc[i][j] = __builtin_amdgcn_wmma_f32_16x16x4_f32(
              false, af[i], false, bf[j], (short)0, acc[i][j], false, false);
    }

    wait_async0();      // my chunk-(c+1) async writes have landed in LDS
    __syncthreads();    // everyone done reading buf and done staging buf^1
  }

  const int mbase = blockIdx.y * BM + m0w + 8 * half;
  const int nbase = blockIdx.x * BN + n0w + l16;
  #pragma unroll
  for (int i = 0; i < 4; ++i) {
    float* cr = Cb + (size_t)(mbase + 16 * i) * ldc + nbase;
    #pragma unroll
    for (int j = 0; j < 4; ++j)
      #pragma unroll
      for (int r = 0; r < 8; ++r)
        cr[(size_t)r * ldc + 16 * j] = acc[i][j][r] * scale;
  }
}

// ---------------------------------------------------------------------------
// NN GEMM: C[m,n] = sum_k A[m,k] * B[k,n]   (ctx = attn @ Vh, N = 64)
// Block 128 threads (4 waves stacked in M), block tile 256x64,
// wave tile 64x64 = 4 M-frags x 4 N-frags (one wave spans all of N).
// ---------------------------------------------------------------------------
__global__ void __launch_bounds__(128) gemm_nn_f32(
    const float* __restrict__ A, const float* __restrict__ B,
    float* __restrict__ C,
    int K, int lda, int ldb, int ldc,
    unsigned long long sA, unsigned long long sB, unsigned long long sC)
{
  const float* Ab = A + (size_t)blockIdx.z * sA;
  const float* Bb = B + (size_t)blockIdx.z * sB;
  float*       Cb = C + (size_t)blockIdx.z * sC;

  const int lane = threadIdx.x & 31;
  const int wave = threadIdx.x >> 5;          // 0..3 -> M
  const int m0   = blockIdx.y * 256 + wave * 64;
  const int half = lane >> 4;
  const int l16  = lane & 15;

  const float* a0 = Ab + (size_t)(m0 + l16) * lda + 2 * half;
  const float* bc = Bb + l16;                 // + 16*j per N-frag, rows strided by ldb

  v8f acc[4][4] = {};
  for (int k = 0; k < K; k += 4) {
    const int kk = k + 2 * half;
    v2f af[4], bf[4];
    #pragma unroll
    for (int i = 0; i < 4; ++i)
      af[i] = *(const v2f*)(a0 + (size_t)(16 * i) * lda + k);
    #pragma unroll
    for (int j = 0; j < 4; ++j) {
      bf[j].x = bc[(size_t)kk * ldb + 16 * j];
      bf[j].y = bc[(size_t)(kk + 1) * ldb + 16 * j];
    }
    #pragma unroll
    for (int i = 0; i < 4; ++i)
      #pragma unroll
      for (int j = 0; j < 4; ++j)
        acc[i][j] = __builtin_amdgcn_wmma_f32_16x16x4_f32(
            false, af[i], false, bf[j], (short)0, acc[i][j], false, false);
  }

  const int mbase = m0 + 8 * half;
  #pragma unroll
  for (int i = 0; i < 4; ++i) {
    float* cr = Cb + (size_t)(mbase + 16 * i) * ldc + l16;
    #pragma unroll
    for (int j = 0; j < 4; ++j)
      #pragma unroll
      for (int r = 0; r < 8; ++r)
        cr[(size_t)r * ldc + 16 * j] = acc[i][j][r];
  }
}

// ---------------------------------------------------------------------------
// Row softmax (in place). One 256-thread block per 4096-wide row; 16 elements
// per thread kept in registers. mask[col] != 0 -> -inf before softmax.
// ---------------------------------------------------------------------------
__global__ void __launch_bounds__(256) softmax_rows(
    float* __restrict__ attn, const unsigned char* __restrict__ mask)
{
  float* row = attn + (size_t)blockIdx.x * S_LEN;
  const int tid = threadIdx.x;

  float v[16];
  float mx = -INFINITY;
  #pragma unroll
  for (int i = 0; i < 16; ++i) {
    const int c = tid + i * 256;
    float x = row[c];
    if (mask[c]) x = -INFINITY;
    v[i] = x;
    mx = fmaxf(mx, x);
  }

  __shared__ float red[256];
  red[tid] = mx;
  __syncthreads();
  #pragma unroll
  for (int s = 128; s > 0; s >>= 1) {
    if (tid < s) red[tid] = fmaxf(red[tid], red[tid + s]);
    __syncthreads();
  }
  mx = red[0];
  __syncthreads();

  float sum = 0.0f;
  #pragma unroll
  for (int i = 0; i < 16; ++i) {
    v[i] = __expf(v[i] - mx);
    sum += v[i];
  }
  red[tid] = sum;
  __syncthreads();
  #pragma unroll
  for (int s = 128; s > 0; s >>= 1) {
    if (tid < s) red[tid] += red[tid + s];
    __syncthreads();
  }
  const float inv = 1.0f / red[0];

  #pragma unroll
  for (int i = 0; i < 16; ++i) row[tid + i * 256] = v[i] * inv;
}

// ---------------------------------------------------------------------------
// Host-side orchestration
// ---------------------------------------------------------------------------
extern "C" void kernel_launch(void* const* d_in, const int* in_sizes, int n_in,
                              void* d_out, int out_size, void* d_ws, size_t ws_size,
                              hipStream_t stream) {
  (void)in_sizes; (void)n_in; (void)out_size; (void)ws_size;

  const float* query = (const float*)d_in[0];
  const float* key   = (const float*)d_in[1];
  const float* value = (const float*)d_in[2];
  const unsigned char* mask = (const unsigned char*)d_in[3];
  const float* w_q = (const float*)d_in[4];
  const float* w_k = (const float*)d_in[5];
  const float* w_v = (const float*)d_in[6];
  const float* w_o = (const float*)d_in[7];

  const size_t SD = (size_t)S_LEN * D_MODEL;        // 3,145,728
  float* out  = (float*)d_out;                      // (S, 768)
  float* attn = out + SD;                           // (12, 4096, 4096)

  float* Qp  = (float*)d_ws;                        // (S, 768) each
  float* Kp  = Qp + SD;
  float* Vp  = Kp + SD;
  float* ctx = Vp + SD;

  const dim3 gProj(D_MODEL / BN, S_LEN / BM, 1);            // (3, 32)
  const dim3 gScore(S_LEN / BN, S_LEN / BM, NUM_HEADS);     // (16, 32, 12)
  const dim3 gCtx(1, S_LEN / 256, NUM_HEADS);               // (1, 16, 12)

  // 1) Projections: P = X @ W^T
  gemm_nt_lds<<<gProj, dim3(256), 0, stream>>>(query, w_q, Qp, D_MODEL, D_MODEL, D_MODEL, D_MODEL, 0ULL, 0ULL, 0ULL, 1.0f);
  gemm_nt_lds<<<gProj, dim3(256), 0, stream>>>(key,   w_k, Kp, D_MODEL, D_MODEL, D_MODEL, D_MODEL, 0ULL, 0ULL, 0ULL, 1.0f);
  gemm_nt_lds<<<gProj, dim3(256), 0, stream>>>(value, w_v, Vp, D_MODEL, D_MODEL, D_MODEL, D_MODEL, 0ULL, 0ULL, 0ULL, 1.0f);

  // 2) scores[h] = (Qh @ Kh^T) / sqrt(64) -> written straight into attn output
  gemm_nt_lds<<<gScore, dim3(256), 0, stream>>>(Qp, Kp, attn, D_KK, D_MODEL, D_MODEL, S_LEN,
                                                (unsigned long long)D_KK, (unsigned long long)D_KK,
                                                (unsigned long long)S_LEN * S_LEN, 0.125f);

  // 3) softmax over each of the 12*4096 rows (in place in d_out)
  softmax_rows<<<dim3(NUM_HEADS * S_LEN), dim3(256), 0, stream>>>(attn, mask);

  // 4) ctx[h] = attn[h] @ Vh   (NN GEMM)
  gemm_nn_f32<<<gCtx, dim3(128), 0, stream>>>(attn, Vp, ctx, S_LEN, S_LEN, D_MODEL, D_MODEL,
                                              (unsigned long long)S_LEN * S_LEN,
                                              (unsigned long long)D_KK, (unsigned long long)D_KK);

  // 5) out = ctx @ w_o^T
  gemm_nt_lds<<<gProj, dim3(256), 0, stream>>>(ctx, w_o, out, D_MODEL, D_MODEL, D_MODEL, D_MODEL, 0ULL, 0ULL, 0ULL, 1.0f);
}